// PhotometricLoss_52862457479182
// MI455X (gfx1250) — compile-verified
//
#include <hip/hip_runtime.h>

#define BATCH 32
#define CH    3
#define HIMG  512
#define WIMG  512
#define NPIX  ((long long)BATCH * CH * HIMG * WIMG)

typedef __attribute__((ext_vector_type(2))) float v2f;
typedef __attribute__((ext_vector_type(8))) float v8f;

// ---------------- Kernel 1: closed-form 3x3 inverses (64 matrices) ----------
__global__ void hinv_kernel(const float* __restrict__ Hp,
                            const float* __restrict__ Ht,
                            float* __restrict__ hinv /* 64*9 floats */) {
    int t = threadIdx.x;
    if (t >= 64) return;
    const float* src = (t < 32) ? (Hp + t * 9) : (Ht + (t - 32) * 9);
    double a00 = src[0], a01 = src[1], a02 = src[2];
    double a10 = src[3], a11 = src[4], a12 = src[5];
    double a20 = src[6], a21 = src[7], a22 = src[8];
    double c00 = a11 * a22 - a12 * a21;
    double c01 = a12 * a20 - a10 * a22;
    double c02 = a10 * a21 - a11 * a20;
    double id  = 1.0 / (a00 * c00 + a01 * c01 + a02 * c02);
    float* o = hinv + t * 9;
    o[0] = (float)(c00 * id);
    o[1] = (float)((a02 * a21 - a01 * a22) * id);
    o[2] = (float)((a01 * a12 - a02 * a11) * id);
    o[3] = (float)(c01 * id);
    o[4] = (float)((a00 * a22 - a02 * a20) * id);
    o[5] = (float)((a02 * a10 - a00 * a12) * id);
    o[6] = (float)(c02 * id);
    o[7] = (float)((a01 * a20 - a00 * a21) * id);
    o[8] = (float)((a00 * a11 - a01 * a10) * id);
}

// ---------------- Per-pixel sampling parameters -----------------------------
struct Samp {
    int   ix0, ix1, iy0, iy1;
    bool  vx0, vx1, vy0, vy1;
    float wx0, wx1, wy0, wy1;
};

__device__ __forceinline__ Samp make_samp(const float h[9], float x, float y) {
    float X  = h[0] * x + h[1] * y + h[2];
    float Y  = h[3] * x + h[4] * y + h[5];
    float Wd = h[6] * x + h[7] * y + h[8];
    float iw = 1.0f / Wd;
    // ((2*X/(W-1) - 1 + 1) * W - 1) * 0.5  ==  X * W/(W-1) - 0.5
    const float scale = (float)WIMG / (float)(WIMG - 1);
    float sx = X * iw * scale - 0.5f;
    float sy = Y * iw * scale - 0.5f;
    float fx0 = floorf(sx), fy0 = floorf(sy);
    float fx1 = fx0 + 1.0f, fy1 = fy0 + 1.0f;
    Samp s;
    s.wx1 = sx - fx0; s.wx0 = 1.0f - s.wx1;
    s.wy1 = sy - fy0; s.wy0 = 1.0f - s.wy1;
    // validity in float space (coords can be astronomically large near w==0)
    s.vx0 = (fx0 >= 0.0f) && (fx0 < (float)WIMG);
    s.vx1 = (fx1 >= 0.0f) && (fx1 < (float)WIMG);
    s.vy0 = (fy0 >= 0.0f) && (fy0 < (float)HIMG);
    s.vy1 = (fy1 >= 0.0f) && (fy1 < (float)HIMG);
    // clamp BEFORE int conversion (avoids UB on huge floats)
    s.ix0 = (int)fminf(fmaxf(fx0, 0.0f), (float)(WIMG - 1));
    s.ix1 = (int)fminf(fmaxf(fx1, 0.0f), (float)(WIMG - 1));
    s.iy0 = (int)fminf(fmaxf(fy0, 0.0f), (float)(HIMG - 1));
    s.iy1 = (int)fminf(fmaxf(fy1, 0.0f), (float)(HIMG - 1));
    return s;
}

__device__ __forceinline__ float sample_bilinear(const float* __restrict__ img,
                                                 const Samp& s) {
    float v00 = (s.vx0 && s.vy0) ? img[s.iy0 * WIMG + s.ix0] : 0.0f;
    float v10 = (s.vx1 && s.vy0) ? img[s.iy0 * WIMG + s.ix1] : 0.0f;
    float v01 = (s.vx0 && s.vy1) ? img[s.iy1 * WIMG + s.ix0] : 0.0f;
    float v11 = (s.vx1 && s.vy1) ? img[s.iy1 * WIMG + s.ix1] : 0.0f;
    return v00 * (s.wx0 * s.wy0) + v10 * (s.wx1 * s.wy0)
         + v01 * (s.wx0 * s.wy1) + v11 * (s.wx1 * s.wy1);
}

// ---------------- Kernel 2: warp-both + squared diff + WMMA wave reduce -----
// grid = (HIMG rows, BATCH), block = 256 (8 wave32), 2 pixels/thread.
__global__ __launch_bounds__(256)
void photoloss_main(const float* __restrict__ I,
                    const float* __restrict__ hinv,
                    float* __restrict__ partials) {
    const int y = blockIdx.x;   // 0..511
    const int b = blockIdx.y;   // 0..31
    const int t = threadIdx.x;  // 0..255

    float hp[9], ht[9];
    const float* __restrict__ hpp = hinv + b * 9;
    const float* __restrict__ htp = hinv + (BATCH + b) * 9;
#pragma unroll
    for (int i = 0; i < 9; ++i) { hp[i] = hpp[i]; ht[i] = htp[i]; }

    const float* __restrict__ Ib = I + (size_t)b * CH * HIMG * WIMG;

    float loss = 0.0f;
    const float fy = (float)y;
#pragma unroll
    for (int k = 0; k < 2; ++k) {
        const float fx = (float)(t + 256 * k);
        Samp sp = make_samp(hp, fx, fy);
        Samp st = make_samp(ht, fx, fy);
#pragma unroll
        for (int c = 0; c < CH; ++c) {
            const float* __restrict__ Ic = Ib + (size_t)c * HIMG * WIMG;
            float d = sample_bilinear(Ic, sp) - sample_bilinear(Ic, st);
            loss += d * d;
        }
    }

    // ---- wave32 sum via V_WMMA_F32_16X16X4_F32 with B = ones --------------
    // A (16x4 f32): lane m v0 -> A[m][0]=s_m (m<16), lane m+16 v0 -> A[m][2]=s_{m+16}
    // D[m][n] = s_m + s_{m+16}; lane L<16 holds rows 0..7 in its 8 D VGPRs,
    // lanes 16..31 hold rows 8..15. Sum 8 VGPRs + xor-16 shuffle = full wave sum.
    v2f a;  a[0] = loss;  a[1] = 0.0f;
    v2f bm; bm[0] = 1.0f; bm[1] = 1.0f;
    v8f cz = {};
    v8f dm = __builtin_amdgcn_wmma_f32_16x16x4_f32(
        /*neg_a=*/false, a, /*neg_b=*/false, bm,
        /*c_mod=*/(short)0, cz, /*reuse_a=*/false, /*reuse_b=*/false);
    float r = dm[0] + dm[1] + dm[2] + dm[3] + dm[4] + dm[5] + dm[6] + dm[7];
    r += __shfl_xor(r, 16, 32);

    __shared__ float wsum[8];
    const int wave = t >> 5, lane = t & 31;
    if (lane == 0) wsum[wave] = r;
    __syncthreads();
    if (t == 0) {
        float s = 0.0f;
#pragma unroll
        for (int w = 0; w < 8; ++w) s += wsum[w];
        partials[b * HIMG + y] = s;   // 16384 block partials, fixed order
    }
}

// ---------------- Kernel 3: deterministic tree reduction -> mean ------------
__global__ __launch_bounds__(1024)
void reduce_partials(const float* __restrict__ partials, float* __restrict__ out) {
    __shared__ double sh[1024];
    const int t = threadIdx.x;
    double s = 0.0;
#pragma unroll
    for (int k = 0; k < 16; ++k) s += (double)partials[t + k * 1024];
    sh[t] = s;
    __syncthreads();
    for (int off = 512; off > 0; off >>= 1) {
        if (t < off) sh[t] += sh[t + off];
        __syncthreads();
    }
    if (t == 0) out[0] = (float)(sh[0] / (double)NPIX);
}

// ---------------- Launch ----------------------------------------------------
extern "C" void kernel_launch(void* const* d_in, const int* in_sizes, int n_in,
                              void* d_out, int out_size, void* d_ws, size_t ws_size,
                              hipStream_t stream) {
    const float* Hp = (const float*)d_in[0];
    const float* Ht = (const float*)d_in[1];
    const float* I  = (const float*)d_in[2];

    float* hinv     = (float*)d_ws;                     // 64*9 floats
    float* partials = (float*)((char*)d_ws + 4096);     // 16384 floats

    hinv_kernel<<<1, 64, 0, stream>>>(Hp, Ht, hinv);

    dim3 grid(HIMG, BATCH);
    photoloss_main<<<grid, 256, 0, stream>>>(I, hinv, partials);

    reduce_partials<<<1, 1024, 0, stream>>>(partials, (float*)d_out);
}